// ConvolutionHebbianLayer_13795434954854
// MI455X (gfx1250) — compile-verified
//
#include <hip/hip_runtime.h>
#include <cmath>

typedef float v2f __attribute__((ext_vector_type(2)));
typedef float v8f __attribute__((ext_vector_type(8)));

namespace {
constexpr int IN_SIZE = 224;
constexpr int IN_CH   = 16;
constexpr int OUT_CH  = 64;
constexpr int NH      = 220;          // (224-5)/1+1
constexpr int NBT     = NH * NH;      // 48400
constexpr int KDIM    = 400;          // 16*5*5
constexpr float LR_OVER_NB = 0.005f / 48400.0f;
constexpr float GAMMA_ = 0.99f;
constexpr float EPSC   = 0.01f;

// workspace layout in floats
constexpr size_t WS_CONV = 0;                    // 48400*64 = 3097600 (dead after k2)
constexpr size_t WS_P1   = 0;                    // yty partials: 16*110*256 = 450560
constexpr size_t WS_P2   = 450560;               // G partials: 100*44*256 = 1126400
constexpr size_t WS_YTY  = 3097600;              // 64*64
constexpr size_t WS_G    = WS_YTY + 4096;        // 64*400
constexpr size_t WS_CSUM = WS_G + 25600;         // 64
constexpr size_t WS_GF   = WS_CSUM + 64;         // 64

constexpr int S1 = 110, KS1 = 440;    // yty K-split: 110*440 = 48400
constexpr int S2 = 44,  KS2 = 1100;   // G   K-split: 44*1100 = 48400
}

// ---------------------------------------------------------------------------
// K1: conv as implicit GEMM  conv[t][o] = sum_k xf[t][k] * Wf[o][k]
// one wave per 16-row M-tile, 4 WMMA accumulators cover all 64 out channels
// ---------------------------------------------------------------------------
__global__ __launch_bounds__(256) void k1_conv(const float* __restrict__ x,
                                               const float* __restrict__ W,
                                               float* __restrict__ conv) {
  int wave = (int)((blockIdx.x * blockDim.x + threadIdx.x) >> 5);
  int lane = (int)(threadIdx.x & 31);
  if (wave >= NBT / 16) return;
  int row0 = wave * 16;
  int lm   = lane & 15;
  int koff = (lane >> 4) << 1;             // K sub-offset 0 or 2

  // fixed spatial position for this lane's A fragment rows
  int t  = row0 + lm;
  int oh = t / NH, ow = t % NH;

  v8f acc0 = {}, acc1 = {}, acc2 = {}, acc3 = {};
  for (int k0 = 0; k0 < KDIM; k0 += 4) {
    int ka = k0 + koff;
    int c1 = ka / 25, re1 = ka % 25, r1 = re1 / 5, s1 = re1 % 5;
    int kb = ka + 1;
    int c2 = kb / 25, re2 = kb % 25, r2 = re2 / 5, s2 = re2 % 5;
    v2f a;
    a.x = x[(c1 * IN_SIZE + (oh + r1)) * IN_SIZE + (ow + s1)];
    a.y = x[(c2 * IN_SIZE + (oh + r2)) * IN_SIZE + (ow + s2)];

    const float* wp = W + lm * KDIM + ka;  // Wf row-major: contiguous float2
    v2f b0 = { wp[0],            wp[1]            };
    v2f b1 = { wp[16 * KDIM],    wp[16 * KDIM + 1] };
    v2f b2 = { wp[32 * KDIM],    wp[32 * KDIM + 1] };
    v2f b3 = { wp[48 * KDIM],    wp[48 * KDIM + 1] };

    acc0 = __builtin_amdgcn_wmma_f32_16x16x4_f32(false, a, false, b0, (short)0, acc0, false, false);
    acc1 = __builtin_amdgcn_wmma_f32_16x16x4_f32(false, a, false, b1, (short)0, acc1, false, false);
    acc2 = __builtin_amdgcn_wmma_f32_16x16x4_f32(false, a, false, b2, (short)0, acc2, false, false);
    acc3 = __builtin_amdgcn_wmma_f32_16x16x4_f32(false, a, false, b3, (short)0, acc3, false, false);
  }

  int mbase = row0 + ((lane >> 4) << 3);   // M = v + 8*(lane>=16)
  for (int v = 0; v < 8; ++v) {
    int trow = mbase + v;
    conv[trow * 64 +      lm] = acc0[v];
    conv[trow * 64 + 16 + lm] = acc1[v];
    conv[trow * 64 + 32 + lm] = acc2[v];
    conv[trow * 64 + 48 + lm] = acc3[v];
  }
}

// ---------------------------------------------------------------------------
// K2: inhibition (relu, /rowmax, ^5) + transpose to NCHW in d_out (== y^T)
// 80 spatial rows per block, LDS transpose
// ---------------------------------------------------------------------------
__global__ __launch_bounds__(256) void k2_inhib(const float* __restrict__ conv,
                                                float* __restrict__ yT) {
  __shared__ float tile[80 * 65];
  __shared__ float rowmax[80];
  int t0  = (int)blockIdx.x * 80;
  int tid = (int)threadIdx.x;

  for (int i = 0; i < 20; ++i) {           // 20*256 = 5120 = 80*64
    int lin = tid + 256 * i;
    int tl = lin >> 6, c = lin & 63;
    float v = conv[(t0 + tl) * 64 + c];
    tile[tl * 65 + c] = v > 0.f ? v : 0.f;
  }
  __syncthreads();
  if (tid < 80) {
    float m = 0.f;
    for (int c = 0; c < 64; ++c) m = fmaxf(m, tile[tid * 65 + c]);
    rowmax[tid] = m + 1e-9f;
  }
  __syncthreads();
  for (int i = 0; i < 20; ++i) {
    int lin = tid + 256 * i;
    int c = lin / 80, tl = lin % 80;
    float v  = tile[tl * 65 + c] / rowmax[tl];
    float v2 = v * v;
    yT[c * NBT + t0 + tl] = v2 * v2 * v;   // v^5
  }
}

// ---------------------------------------------------------------------------
// K4: column sums of y  (mean over axis 0) — deterministic tree reduction
// ---------------------------------------------------------------------------
__global__ __launch_bounds__(256) void k4_colsum(const float* __restrict__ yT,
                                                 float* __restrict__ colsum) {
  int c = (int)blockIdx.x, tid = (int)threadIdx.x;
  float s = 0.f;
  for (int t = tid; t < NBT; t += 256) s += yT[c * NBT + t];
  __shared__ float red[256];
  red[tid] = s; __syncthreads();
  for (int off = 128; off > 0; off >>= 1) {
    if (tid < off) red[tid] += red[tid + off];
    __syncthreads();
  }
  if (tid == 0) colsum[c] = red[0];
}

// ---------------------------------------------------------------------------
// K5: yty partials — yty[i][j] = sum_t yT[i][t]*yT[j][t], K split into S1 slices
// ---------------------------------------------------------------------------
__global__ __launch_bounds__(256) void k5_yty_part(const float* __restrict__ yT,
                                                   float* __restrict__ part) {
  int gw   = (int)((blockIdx.x * blockDim.x + threadIdx.x) >> 5);
  int lane = (int)(threadIdx.x & 31);
  if (gw >= 16 * S1) return;
  int tp = gw / S1, s = gw % S1;
  int i0 = (tp >> 2) * 16, j0 = (tp & 3) * 16;
  int lm = lane & 15;
  int toff = (lane >> 4) << 1;
  const float* arow = yT + (i0 + lm) * NBT;
  const float* brow = yT + (j0 + lm) * NBT;
  v8f acc = {};
  int tbase = s * KS1;
  for (int kk = 0; kk < KS1; kk += 4) {
    int t = tbase + kk + toff;
    v2f a = { arow[t], arow[t + 1] };
    v2f b = { brow[t], brow[t + 1] };
    acc = __builtin_amdgcn_wmma_f32_16x16x4_f32(false, a, false, b, (short)0, acc, false, false);
  }
  float* p = part + (size_t)(tp * S1 + s) * 256;
  for (int v = 0; v < 8; ++v) p[v * 32 + lane] = acc[v];
}

__global__ __launch_bounds__(256) void k5b_yty_red(const float* __restrict__ part,
                                                   float* __restrict__ yty) {
  int tp = (int)blockIdx.x, e = (int)threadIdx.x;
  float sum = 0.f;
  for (int s = 0; s < S1; ++s) sum += part[(size_t)(tp * S1 + s) * 256 + e];
  int v = e >> 5, lane = e & 31;
  int i = (tp >> 2) * 16 + v + ((lane >> 4) << 3);
  int j = (tp & 3) * 16 + (lane & 15);
  yty[i * 64 + j] = sum;
}

// ---------------------------------------------------------------------------
// K7: G partials — G[c][k] = sum_t yT[c][t]*xf[t][k], K split into S2 slices
// ---------------------------------------------------------------------------
__global__ __launch_bounds__(256) void k7_g_part(const float* __restrict__ yT,
                                                 const float* __restrict__ x,
                                                 float* __restrict__ part) {
  int gw   = (int)((blockIdx.x * blockDim.x + threadIdx.x) >> 5);
  int lane = (int)(threadIdx.x & 31);
  if (gw >= 100 * S2) return;
  int tp = gw / S2, s = gw % S2;
  int mt = tp / 25, nt = tp % 25;
  int c0 = mt * 16, k0 = nt * 16;
  int lm = lane & 15;
  int toff = (lane >> 4) << 1;

  // fixed patch coordinate for this lane's B column
  int kk = k0 + lm;
  int ci = kk / 25, rem = kk % 25, r = rem / 5, ss = rem % 5;
  const float* xb = x + (ci * IN_SIZE + r) * IN_SIZE + ss;
  const float* arow = yT + (c0 + lm) * NBT;

  v8f acc = {};
  int tbase = s * KS2;
  for (int step = 0; step < KS2; step += 4) {
    int t = tbase + step + toff;
    v2f a = { arow[t], arow[t + 1] };
    int oh1 = t / NH,       ow1 = t % NH;
    int oh2 = (t + 1) / NH, ow2 = (t + 1) % NH;
    v2f b = { xb[oh1 * IN_SIZE + ow1], xb[oh2 * IN_SIZE + ow2] };
    acc = __builtin_amdgcn_wmma_f32_16x16x4_f32(false, a, false, b, (short)0, acc, false, false);
  }
  float* p = part + (size_t)(tp * S2 + s) * 256;
  for (int v = 0; v < 8; ++v) p[v * 32 + lane] = acc[v];
}

__global__ __launch_bounds__(256) void k7b_g_red(const float* __restrict__ part,
                                                 float* __restrict__ G) {
  int tp = (int)blockIdx.x, e = (int)threadIdx.x;
  float sum = 0.f;
  for (int s = 0; s < S2; ++s) sum += part[(size_t)(tp * S2 + s) * 256 + e];
  int mt = tp / 25, nt = tp % 25;
  int v = e >> 5, lane = e & 31;
  int i = mt * 16 + v + ((lane >> 4) << 3);
  int j = nt * 16 + (lane & 15);
  G[i * KDIM + j] = sum;
}

// ---------------------------------------------------------------------------
// K6: exp_new, A = exp_new/mean, gf = eps*tanh(-eps*(A-1))+1
// ---------------------------------------------------------------------------
__global__ void k6_expnew(const float* __restrict__ colsum,
                          const float* __restrict__ exp_avg,
                          float* __restrict__ gf,
                          float* __restrict__ expnew_out) {
  int c = (int)threadIdx.x;                // 64 threads
  float en = GAMMA_ * exp_avg[c] + (1.0f - GAMMA_) * (colsum[c] / (float)NBT);
  __shared__ float sh[64];
  sh[c] = en; __syncthreads();
  for (int off = 32; off > 0; off >>= 1) {
    if (c < off) sh[c] += sh[c + off];
    __syncthreads();
  }
  float mean = sh[0] / 64.0f;
  float A = en / mean;
  gf[c] = EPSC * tanhf(-EPSC * (A - 1.0f)) + 1.0f;
  expnew_out[c] = en;
}

// ---------------------------------------------------------------------------
// K8: Wnew = relu(Wf + LR/NB*(G - yty@Wf)); Wfinal = pos*gfp + neg*gfn
// (Wnew >= 0 after relu, so neg part is identically zero)
// ---------------------------------------------------------------------------
__global__ __launch_bounds__(256) void k8_wfinal(const float* __restrict__ W,
                                                 const float* __restrict__ G,
                                                 const float* __restrict__ yty,
                                                 const float* __restrict__ gf,
                                                 float* __restrict__ wout) {
  int idx = (int)(blockIdx.x * blockDim.x + threadIdx.x);
  if (idx >= OUT_CH * KDIM) return;
  int o = idx / KDIM, k = idx % KDIM;
  float s = 0.f;
  for (int j = 0; j < 64; ++j) s += yty[o * 64 + j] * W[j * KDIM + k];
  float wnew = W[idx] + LR_OVER_NB * (G[idx] - s);
  wnew = wnew > 0.f ? wnew : 0.f;
  wout[idx] = wnew * gf[o];
}

// ---------------------------------------------------------------------------
extern "C" void kernel_launch(void* const* d_in, const int* in_sizes, int n_in,
                              void* d_out, int out_size, void* d_ws, size_t ws_size,
                              hipStream_t stream) {
  const float* x       = (const float*)d_in[0];   // (1,16,224,224)
  const float* W       = (const float*)d_in[1];   // (64,16,5,5) == Wf (64,400)
  const float* exp_avg = (const float*)d_in[2];   // (64,)
  (void)in_sizes; (void)n_in; (void)out_size; (void)ws_size;

  float* ws   = (float*)d_ws;
  float* outp = (float*)d_out;              // out NCHW: 64*220*220
  float* wfin = outp + (size_t)OUT_CH * NBT;        // Wfinal: 64*400
  float* expn = wfin + (size_t)OUT_CH * KDIM;       // exp_new: 64

  float* conv   = ws + WS_CONV;
  float* p1     = ws + WS_P1;
  float* p2     = ws + WS_P2;
  float* yty    = ws + WS_YTY;
  float* G      = ws + WS_G;
  float* colsum = ws + WS_CSUM;
  float* gf     = ws + WS_GF;

  // conv (WMMA implicit GEMM): 3025 waves
  k1_conv<<<(3025 + 7) / 8, 256, 0, stream>>>(x, W, conv);
  // inhibition + transpose into d_out (y^T, NCHW)
  k2_inhib<<<NBT / 80, 256, 0, stream>>>(conv, outp);
  // column sums for mean(y, axis=0)
  k4_colsum<<<64, 256, 0, stream>>>(outp, colsum);
  // yty = y^T y (WMMA, K-split + deterministic reduce)
  k5_yty_part<<<(16 * S1 * 32) / 256, 256, 0, stream>>>(outp, p1);
  k5b_yty_red<<<16, 256, 0, stream>>>(p1, yty);
  // G = y^T xf (WMMA, K-split + deterministic reduce)
  k7_g_part<<<(100 * S2 * 32) / 256, 256, 0, stream>>>(outp, x, p2);
  k7b_g_red<<<100, 256, 0, stream>>>(p2, G);
  // exp_new + gating factors
  k6_expnew<<<1, 64, 0, stream>>>(colsum, exp_avg, gf, expn);
  // final weight update
  k8_wfinal<<<100, 256, 0, stream>>>(W, G, yty, gf, wfin);
}